// PointNetFeaturePropagation_87076166960236
// MI455X (gfx1250) — compile-verified
//
#include <hip/hip_runtime.h>
#include <hip/hip_bf16.h>

typedef __attribute__((ext_vector_type(2))) float v2f;
typedef __attribute__((ext_vector_type(8))) float v8f;
typedef __attribute__((ext_vector_type(4))) unsigned int u32x4;
typedef __attribute__((ext_vector_type(4))) int i32x4;
typedef __attribute__((ext_vector_type(8))) int i32x8;

#define N_PTS 16384
#define S_PTS 2048
#define D1 128
#define D2 256
#define MLP0 256
#define MLP1 128
#define KDIM (D1 + D2)   // 384
#define BN_EPS 1e-5f

#define BM 32            // points (M rows) per block in the MLP kernel
#define FEAT_LD 385      // odd float stride -> conflict-free column reads
#define HID_LD 257

// ---------------------------------------------------------------------------
// Kernel 1: 3-nearest-neighbor + normalized inverse-distance weights.
// xyz2 ([3][2048] f32, 24 KB contiguous) is staged into LDS via the Tensor
// Data Mover (tensor_load_to_lds + s_wait_tensorcnt). A cooperative ds_store
// path writes the identical bytes so LDS contents are correct regardless of
// TDM descriptor interpretation (same values -> order-independent).
// ---------------------------------------------------------------------------
__global__ __launch_bounds__(256)
void knn3_kernel(const float* __restrict__ xyz1, const float* __restrict__ xyz2,
                 float* __restrict__ w_out, int* __restrict__ idx_out) {
    __shared__ float s2[3 * S_PTS];   // [0,2048)=x, [2048,4096)=y, [4096,6144)=z

#if __has_builtin(__builtin_amdgcn_tensor_load_to_lds)
    if (threadIdx.x == 0) {
        const unsigned long long gaddr = (unsigned long long)xyz2;
        const unsigned lds_off = (unsigned)(unsigned long long)(void*)s2;
        const unsigned elems = 3 * S_PTS;              // 6144 f32, one DMA row

        u32x4 g0;
        g0.x = 1u;                                     // count=1, user D#
        g0.y = lds_off;                                // LDS byte address
        g0.z = (unsigned)(gaddr & 0xffffffffull);      // global_addr[31:0]
        g0.w = (unsigned)((gaddr >> 32) & 0x01ffffffull) | (2u << 30); // type=2

        i32x8 g1;
        g1[0] = (int)(2u << 16);                       // data_size=2 -> 4 bytes
        g1[1] = (int)((elems & 0xffffu) << 16);        // tensor_dim0[15:0] @bits63:48
        g1[2] = (int)(((elems >> 16) & 0xffffu) | (1u << 16)); // dim0 hi | tensor_dim1=1
        g1[3] = (int)((elems & 0xffffu) << 16);        // tile_dim0 @bits127:112
        g1[4] = 1;                                     // tile_dim1=1, tile_dim2=0
        g1[5] = (int)elems;                            // tensor_dim0_stride[31:0]
        g1[6] = (int)((elems & 0xffffu) << 16);        // stride0 hi=0 | stride1[15:0]
        g1[7] = 0;                                     // stride1 hi
        i32x4 gz = {};
#if defined(__clang_major__) && (__clang_major__ >= 23)
        i32x8 gz8 = {};
        __builtin_amdgcn_tensor_load_to_lds(g0, g1, gz, gz, gz8, 0);
#else
        __builtin_amdgcn_tensor_load_to_lds(g0, g1, gz, gz, 0);
#endif
        __builtin_amdgcn_s_wait_tensorcnt(0);
    }
#endif
    // Cooperative copy of the same 24 KB (belt-and-suspenders; identical data)
    for (int s = threadIdx.x; s < 3 * S_PTS; s += 256) s2[s] = xyz2[s];
    __syncthreads();

    const float* sx = s2;
    const float* sy = s2 + S_PTS;
    const float* sz = s2 + 2 * S_PTS;

    const int n = blockIdx.x * 256 + threadIdx.x;
    const float x = xyz1[0 * N_PTS + n];
    const float y = xyz1[1 * N_PTS + n];
    const float z = xyz1[2 * N_PTS + n];

    float d0 = 3.4e38f, d1 = 3.4e38f, d2 = 3.4e38f;
    int   i0 = 0,       i1 = 0,       i2 = 0;
    for (int s = 0; s < S_PTS; ++s) {
        const float dx = x - sx[s], dy = y - sy[s], dz = z - sz[s];
        const float d  = dx * dx + dy * dy + dz * dz;
        const bool c0 = d < d0, c1 = d < d1, c2 = d < d2;
        d2 = c1 ? d1 : (c2 ? d : d2);
        i2 = c1 ? i1 : (c2 ? s : i2);
        d1 = c0 ? d0 : (c1 ? d : d1);
        i1 = c0 ? i0 : (c1 ? s : i1);
        d0 = c0 ? d  : d0;
        i0 = c0 ? s  : i0;
    }
    const float w0 = 1.0f / (d0 + 1e-8f);
    const float w1 = 1.0f / (d1 + 1e-8f);
    const float w2 = 1.0f / (d2 + 1e-8f);
    const float inv = 1.0f / (w0 + w1 + w2);
    w_out[n * 3 + 0] = w0 * inv;
    w_out[n * 3 + 1] = w1 * inv;
    w_out[n * 3 + 2] = w2 * inv;
    idx_out[n * 3 + 0] = i0;
    idx_out[n * 3 + 1] = i1;
    idx_out[n * 3 + 2] = i2;
}

// ---------------------------------------------------------------------------
// Kernel 2: fused interpolate+concat -> dense(384,256)+BN+ReLU ->
//           dense(256,128)+BN+ReLU, all via V_WMMA_F32_16X16X4_F32
// ---------------------------------------------------------------------------
__global__ __launch_bounds__(256)
void fp_mlp_kernel(const float* __restrict__ points1, const float* __restrict__ points2,
                   const float* __restrict__ w3, const int* __restrict__ idx3,
                   const float* __restrict__ W0, const float* __restrict__ b0,
                   const float* __restrict__ g0, const float* __restrict__ be0,
                   const float* __restrict__ m0, const float* __restrict__ v0,
                   const float* __restrict__ W1, const float* __restrict__ b1,
                   const float* __restrict__ g1, const float* __restrict__ be1,
                   const float* __restrict__ m1, const float* __restrict__ v1,
                   float* __restrict__ out) {
    // One buffer: feat tile [BM][384] (stride 385) reused as hidden [BM][256] (stride 257)
    __shared__ float smem[BM * FEAT_LD];   // 49280 bytes

    const int tid  = threadIdx.x;
    const int row0 = blockIdx.x * BM;

    // Warm L2 for the weight matrices (global_prefetch_b8)
    __builtin_prefetch(&W0[tid * 64], 0, 1);
    __builtin_prefetch(&W1[tid * 32], 0, 1);

    // ---- build feat tile in LDS: cols [0,128) = points1^T, [128,384) = interp ----
    for (int e = tid; e < BM * KDIM; e += 256) {
        const int r = e / KDIM;
        const int c = e % KDIM;
        const int p = row0 + r;
        float val;
        if (c < D1) {
            val = points1[c * N_PTS + p];
        } else {
            const int d = c - D1;
            const float* prow = &points2[d * S_PTS];
            val  = w3[p * 3 + 0] * prow[idx3[p * 3 + 0]];
            val += w3[p * 3 + 1] * prow[idx3[p * 3 + 1]];
            val += w3[p * 3 + 2] * prow[idx3[p * 3 + 2]];
        }
        smem[r * FEAT_LD + c] = val;
    }
    __syncthreads();

    const int lane = tid & 31;     // wave32
    const int wave = tid >> 5;     // 0..7
    const int lo   = lane & 15;
    const int hi   = lane >> 4;

    // ================= layer 0: [BM x 384] @ [384 x 256] =================
    // 2 M-subtiles x 16 N-subtiles = 32 tiles; wave owns (m_sub, 4 N-subtiles)
    const int m_sub = wave & 1;
    const int n0    = (wave >> 1) * 4;

    v8f acc[4] = {};
    for (int k = 0; k < KDIM; k += 4) {
        // A fragment: lane l, vgpr j -> feat[m_sub*16 + lo][k + 2*hi + j]
        const float* arow = &smem[(m_sub * 16 + lo) * FEAT_LD + k + 2 * hi];
        v2f afrag = { arow[0], arow[1] };
#pragma unroll
        for (int t = 0; t < 4; ++t) {
            const int nb = (n0 + t) * 16;
            // B fragment: vgpr j -> W0[k + j + 2*hi][nb + lo]
            const float* bp = &W0[(k + 2 * hi) * MLP0 + nb + lo];
            v2f bfrag = { bp[0], bp[MLP0] };
            acc[t] = __builtin_amdgcn_wmma_f32_16x16x4_f32(
                false, afrag, false, bfrag, (short)0, acc[t], false, false);
        }
    }
    __syncthreads();   // everyone done reading feat; buffer reused for hidden

    // BN (inference) + linear bias folded: z = relu(acc*sc + sh)
#pragma unroll
    for (int t = 0; t < 4; ++t) {
        const int col = (n0 + t) * 16 + lo;
        const float sc = g0[col] * rsqrtf(v0[col] + BN_EPS);
        const float sh = (b0[col] - m0[col]) * sc + be0[col];
#pragma unroll
        for (int r = 0; r < 8; ++r) {
            const int m = m_sub * 16 + r + 8 * hi;       // C/D row layout
            const float h = fmaxf(acc[t][r] * sc + sh, 0.0f);
            smem[m * HID_LD + col] = h;
        }
    }
    __syncthreads();

    // ================= layer 1: [BM x 256] @ [256 x 128] =================
    // 2 M-subtiles x 8 N-subtiles = 16 tiles; wave owns (m_sub, 2 N-subtiles)
    const int n1 = (wave >> 1) * 2;
    v8f acc1[2] = {};
    for (int k = 0; k < MLP0; k += 4) {
        const float* arow = &smem[(m_sub * 16 + lo) * HID_LD + k + 2 * hi];
        v2f afrag = { arow[0], arow[1] };
#pragma unroll
        for (int t = 0; t < 2; ++t) {
            const int nb = (n1 + t) * 16;
            const float* bp = &W1[(k + 2 * hi) * MLP1 + nb + lo];
            v2f bfrag = { bp[0], bp[MLP1] };
            acc1[t] = __builtin_amdgcn_wmma_f32_16x16x4_f32(
                false, afrag, false, bfrag, (short)0, acc1[t], false, false);
        }
    }

    // BN + ReLU + transposed store: out[col * N + point]
#pragma unroll
    for (int t = 0; t < 2; ++t) {
        const int col = (n1 + t) * 16 + lo;
        const float sc = g1[col] * rsqrtf(v1[col] + BN_EPS);
        const float sh = (b1[col] - m1[col]) * sc + be1[col];
#pragma unroll
        for (int r = 0; r < 8; ++r) {
            const int p = row0 + m_sub * 16 + r + 8 * hi;
            out[col * N_PTS + p] = fmaxf(acc1[t][r] * sc + sh, 0.0f);
        }
    }
}

// ---------------------------------------------------------------------------
extern "C" void kernel_launch(void* const* d_in, const int* in_sizes, int n_in,
                              void* d_out, int out_size, void* d_ws, size_t ws_size,
                              hipStream_t stream) {
    const float* xyz1    = (const float*)d_in[0];
    const float* xyz2    = (const float*)d_in[1];
    const float* points1 = (const float*)d_in[2];
    const float* points2 = (const float*)d_in[3];
    const float* W0  = (const float*)d_in[4];
    const float* b0  = (const float*)d_in[5];
    const float* g0  = (const float*)d_in[6];
    const float* be0 = (const float*)d_in[7];
    const float* m0  = (const float*)d_in[8];
    const float* v0  = (const float*)d_in[9];
    const float* W1  = (const float*)d_in[10];
    const float* b1  = (const float*)d_in[11];
    const float* g1  = (const float*)d_in[12];
    const float* be1 = (const float*)d_in[13];
    const float* m1  = (const float*)d_in[14];
    const float* v1  = (const float*)d_in[15];
    float* out = (float*)d_out;

    float* w3   = (float*)d_ws;
    int*   idx3 = (int*)((char*)d_ws + (size_t)N_PTS * 3 * sizeof(float));

    knn3_kernel<<<N_PTS / 256, 256, 0, stream>>>(xyz1, xyz2, w3, idx3);
    fp_mlp_kernel<<<N_PTS / BM, 256, 0, stream>>>(points1, points2, w3, idx3,
                                                  W0, b0, g0, be0, m0, v0,
                                                  W1, b1, g1, be1, m1, v1, out);
}